// LinearTransform_88682484728533
// MI455X (gfx1250) — compile-verified
//
#include <hip/hip_runtime.h>
#include <cstdint>

// Problem constants (from the reference): x is (B, SLOTS) fp32, w is (D, SLOTS) fp32.
#define SLOTS 65536
#define SMASK (SLOTS - 1)
#define NFOLD 16   // 2^i * stride mod 65536 == 0 for i >= 16 -> those diagonals fold onto identity
#define DTOT  32
#define BLOCK 1024 // 32 wave32 waves per workgroup
#define HALF  (SLOTS / 2)

typedef unsigned u32x4 __attribute__((ext_vector_type(4)));
typedef unsigned u32x8 __attribute__((ext_vector_type(8)));

// ---- CDNA5 async global -> LDS copy (ASYNCcnt) -------------------------------
__device__ __forceinline__ void async_copy_b128_to_lds(unsigned lds_byte_addr,
                                                       const void* gaddr) {
  asm volatile("global_load_async_to_lds_b128 %0, %1, off"
               :
               : "v"(lds_byte_addr), "v"((unsigned long long)(uintptr_t)gaddr)
               : "memory");
}

__device__ __forceinline__ void wait_async_zero() {
  asm volatile("s_wait_asynccnt 0" ::: "memory");
}

// ---- CDNA5 Tensor Data Mover: DMA a 16384 x <lines> fp32 tile into LDS -------
// D# layout per CDNA5 ISA (2D tensor, groups 2/3 NULL).
__device__ __forceinline__ void tdm_load_tile_to_lds(const void* gsrc,
                                                     unsigned lds_byte_addr,
                                                     unsigned lines) {
  const unsigned long long ga = (unsigned long long)(uintptr_t)gsrc;
  u32x4 g0;
  g0.x = 1u;                                   // count=1 (valid user D#), gather off
  g0.y = lds_byte_addr;                        // lds_addr (bytes)
  g0.z = (unsigned)ga;                         // global_addr[31:0]
  g0.w = ((unsigned)(ga >> 32) & 0x01FFFFFFu)  // global_addr[56:32]
         | 0x80000000u;                        // type=2 ("image") in bits[127:126]
  u32x8 g1;
  g1.s0 = 2u << 16;                            // workgroup_mask=0, data_size=2 (4 B)
  g1.s1 = 16384u << 16;                        // tensor_dim0 = 16384 (elements)
  g1.s2 = lines << 16;                         // tensor_dim1 = lines
  g1.s3 = 16384u << 16;                        // tile_dim0 = 16384
  g1.s4 = lines;                               // tile_dim1 = lines, tile_dim2 = 0
  g1.s5 = 16384u;                              // tensor_dim0_stride[31:0] = 16384
  g1.s6 = 0u;                                  // dim0_stride[47:32], dim1_stride[15:0]
  g1.s7 = 0u;                                  // tensor_dim1_stride[47:16]
  asm volatile("tensor_load_to_lds %0, %1" : : "s"(g0), "s"(g1) : "memory");
}

__device__ __forceinline__ void wait_tensor_zero() {
  asm volatile("s_wait_tensorcnt 0" ::: "memory");
}

// ---- Pass 1: fold the 16 identity diagonals: wsum[s] = sum_{i=16..31} w[i,s] --
__global__ void wsum16_kernel(const float* __restrict__ w, float* __restrict__ wsum) {
  const int s = blockIdx.x * blockDim.x + threadIdx.x;
  float a = 0.f;
#pragma unroll
  for (int i = NFOLD; i < DTOT; ++i) a += w[(size_t)i * SLOTS + s];
  wsum[s] = a;
}

// ---- Pass 2: one workgroup per row b; stage the whole 256 KiB row in LDS -----
__global__ void __launch_bounds__(BLOCK) bsgs_diag_kernel(
    const float* __restrict__ x, const float* __restrict__ w,
    const float* __restrict__ wsum,   // may be null -> compute fold inline
    const int* __restrict__ stride_p,
    float* __restrict__ out) {
  extern __shared__ float lds_x[];  // SLOTS floats = 256 KiB (<= 320 KB/WGP)

  const unsigned tid = threadIdx.x;
  const unsigned b   = blockIdx.x;
  const float* __restrict__ xrow = x + (size_t)b * SLOTS;
  float* __restrict__       orow = out + (size_t)b * SLOTS;
  const unsigned stride   = (unsigned)stride_p[0];
  const unsigned lds_base = (unsigned)__builtin_amdgcn_groupstaticsize(); // dyn-LDS base

  // --- Stage the row with BOTH CDNA5 copy engines, disjoint halves ---
  // Wave 0: TDM DMAs the first half (32768 floats = 16384 x 2 tile).
  if (tid < 32) {
    tdm_load_tile_to_lds(xrow, lds_base, /*lines=*/2);
  }
  // All 32 waves: async b128 copies for the second half (8 issues / lane).
#pragma unroll
  for (int it = 0; it < HALF / (BLOCK * 4); ++it) {
    const unsigned e = HALF + (unsigned)(it * BLOCK + tid) * 4u;   // float index
    async_copy_b128_to_lds(lds_base + e * 4u, xrow + e);
  }
  wait_async_zero();                 // this wave's async copies landed
  if (tid < 32) wait_tensor_zero();  // wave 0: TDM transfer landed
  __syncthreads();                   // publish LDS to all waves

  // Rotation amounts (uniform; u32 overflow is harmless: 2^32 % 65536 == 0).
  unsigned shifts[NFOLD];
#pragma unroll
  for (int i = 0; i < NFOLD; ++i) shifts[i] = ((1u << i) * stride) & SMASK;

  for (unsigned s = tid; s < SLOTS; s += BLOCK) {
    const float xs = lds_x[s];
    float acc;
    if (wsum) {
      acc = xs * wsum[s];
    } else {  // workspace too small: fold inline (extra L2-resident streaming)
      float wa = 0.f;
#pragma unroll
      for (int i = NFOLD; i < DTOT; ++i) wa += w[(size_t)i * SLOTS + s];
      acc = xs * wa;
    }
#pragma unroll
    for (int i = 0; i < NFOLD; ++i) {
      acc = __builtin_fmaf(lds_x[(s + shifts[i]) & SMASK],   // conflict-free LDS gather
                           w[(size_t)i * SLOTS + s],          // coalesced, L2-resident (8 MiB)
                           acc);
    }
    // Streaming output: keep it out of L2 so the weights stay hot.
    __builtin_nontemporal_store(acc, &orow[s]);
  }
}

extern "C" void kernel_launch(void* const* d_in, const int* in_sizes, int n_in,
                              void* d_out, int out_size, void* d_ws, size_t ws_size,
                              hipStream_t stream) {
  (void)n_in; (void)out_size;
  const float* x        = (const float*)d_in[0];
  const float* w        = (const float*)d_in[1];
  const int*   stride_p = (const int*)d_in[2];
  float*       out      = (float*)d_out;

  const int Brows = in_sizes[0] / SLOTS;   // 512

  float* wsum = nullptr;
  if (ws_size >= (size_t)SLOTS * sizeof(float)) {
    wsum = (float*)d_ws;
    wsum16_kernel<<<SLOTS / 256, 256, 0, stream>>>(w, wsum);
  }

  bsgs_diag_kernel<<<Brows, BLOCK, (size_t)SLOTS * sizeof(float), stream>>>(
      x, w, wsum, stride_p, out);
}